// SPIKE_layer_11647951306858
// MI455X (gfx1250) — compile-verified
//
#include <hip/hip_runtime.h>
#include <cstdint>
#include <cstddef>

// LIF spike scan, MI455X (gfx1250).
// x [B=32, T=8, S=131072] fp32 -> out same shape.
// 268 MB streamed -> ~11.5 us floor @ 23.3 TB/s; pure bandwidth kernel.
//
// Per thread: one float4 of S, all 8 timesteps.
//  - 8x global_load_async_to_lds_b128 (ASYNCcnt, NT hint, no VGPR landing),
//    issued back-to-back for full memory-level parallelism.
//  - pipelined s_wait_asynccnt (T-1-t): loads complete in order, so the scan
//    for timestep t overlaps the in-flight loads for t+1..7.
//  - ds_load_b128 readback (wave-local data only -> no block barrier needed),
//  - NT global_store_b128 outputs (write-once, dataset > 192 MB L2).

#define THREADS 256
#define T_STEPS 8

typedef float f4 __attribute__((ext_vector_type(4)));

template <int S_C, int BPB_C>
__global__ __launch_bounds__(THREADS) void lif_scan_async_kernel(
    const float* __restrict__ x,
    const float* __restrict__ th_inner,   // length-1
    const float* __restrict__ th_outer,   // length-1
    float* __restrict__ out,
    int S_rt, int bpb_rt)
{
    const int S   = (S_C   > 0) ? S_C   : S_rt;    // floats per (b,t)
    const int BPB = (BPB_C > 0) ? BPB_C : bpb_rt;  // blocks per batch index

    __shared__ f4 smem[T_STEPS * THREADS];   // 32 KiB

    const int tid = threadIdx.x;

    // A block never straddles a batch index (S/4 is a multiple of THREADS).
    const int bIdx     = blockIdx.x / BPB;               // shift when BPB_C=128
    const int blockInB = blockIdx.x - bIdx * BPB;
    const long long s4   = (long long)blockInB * THREADS + tid;
    const long long base = ((long long)bIdx * T_STEPS) * (long long)S + s4 * 4;

    // Raw wave-relative LDS byte address of this thread's t=0 slot.
    unsigned lds_base;
    {
        __attribute__((address_space(3))) f4* lp =
            (__attribute__((address_space(3))) f4*)&smem[tid];
        lds_base = (unsigned)(uintptr_t)lp;
    }

    // Fire all 8 async 128-bit global->LDS copies (one per timestep).
#pragma unroll
    for (int t = 0; t < T_STEPS; ++t) {
        const float* gp = x + base + (long long)t * (long long)S;
        unsigned lds_off = lds_base + (unsigned)(t * (THREADS * 16));
        asm volatile("global_load_async_to_lds_b128 %0, %1, off th:TH_LOAD_NT"
                     :
                     : "v"(lds_off), "v"(gp)
                     : "memory");
    }

    // Uniform thresholds (length-1 arrays; Tf = T = 8).
    const float TH   = th_inner[0] * (float)T_STEPS;
    const float EMIT = th_outer[0] * (float)T_STEPS;

    f4 mem = {0.f, 0.f, 0.f, 0.f};
#pragma unroll
    for (int t = 0; t < T_STEPS; ++t) {
        // Async loads complete in order: waiting ASYNCcnt <= 7-t guarantees
        // load t has landed in LDS while later loads stay in flight.
        asm volatile("s_wait_asynccnt %c0" :: "n"(T_STEPS - 1 - t) : "memory");

        f4 v = smem[t * THREADS + tid];           // ds_load_b128, wave-local
        mem.x += v.x; mem.y += v.y; mem.z += v.z; mem.w += v.w;

        f4 sp;
        sp.x = (mem.x >= TH) ? EMIT : 0.0f;
        sp.y = (mem.y >= TH) ? EMIT : 0.0f;
        sp.z = (mem.z >= TH) ? EMIT : 0.0f;
        sp.w = (mem.w >= TH) ? EMIT : 0.0f;

        mem.x -= sp.x; mem.y -= sp.y; mem.z -= sp.z; mem.w -= sp.w;

        f4* op = (f4*)(out + base + (long long)t * (long long)S);
        __builtin_nontemporal_store(sp, op);      // NT global_store_b128
    }
}

extern "C" void kernel_launch(void* const* d_in, const int* in_sizes, int n_in,
                              void* d_out, int out_size, void* d_ws, size_t ws_size,
                              hipStream_t stream) {
    (void)n_in; (void)d_ws; (void)ws_size; (void)out_size;

    const float* x        = (const float*)d_in[0];
    const float* th_inner = (const float*)d_in[1];
    const float* th_outer = (const float*)d_in[2];
    // d_in[3] is the python scalar `t` (unused by the math).
    float* out = (float*)d_out;

    const int total = in_sizes[0];              // B*T*S
    const int B = 32;
    const int S = total / (B * T_STEPS);        // 131072 for the given shape
    const int blocksPerB = (S / 4) / THREADS;   // 128
    const int blocks = B * blocksPerB;          // 4096

    if (S == 131072 && blocksPerB == 128) {
        lif_scan_async_kernel<131072, 128><<<blocks, THREADS, 0, stream>>>(
            x, th_inner, th_outer, out, 0, 0);
    } else {
        lif_scan_async_kernel<0, 0><<<blocks, THREADS, 0, stream>>>(
            x, th_inner, th_outer, out, S, blocksPerB);
    }
}